// ESN_19284403159401
// MI455X (gfx1250) — compile-verified
//
#include <hip/hip_runtime.h>
#include <hip/hip_bf16.h>
#include <math.h>

// ---------------- problem constants ----------------
#define RDIM   1024
#define IDIM   8
#define TSTEPS 65536
#define NWG    32                      // persistent workgroups, rows split 32-way
#define ROWS_PER_WG (RDIM / NWG)       // 32
#define SCAN_THREADS 1024
#define KSEG  (SCAN_THREADS / ROWS_PER_WG)        // 32 k-segments
#define KPER  (RDIM / KSEG)                       // 32 k per segment
#define PF_DIST 8                       // U prefetch distance (steps)

// LDS carve-up (dynamic): Wt + s + red + Win = 140,288 B (< 320 KB/WGP)
#define SMEM_FLOATS (ROWS_PER_WG*RDIM + RDIM + KSEG*ROWS_PER_WG + ROWS_PER_WG*IDIM)
#define SMEM_BYTES  (SMEM_FLOATS * 4)

// workspace layout
#define FLAG_STRIDE 32                 // uints: 128 B per flag line
#define WS_FLAGS_BYTES (NWG * FLAG_STRIDE * 4)    // 4 KB
#define WS_SBUF_OFF  8192
#define WS_U_OFF     16384

typedef float v2f __attribute__((ext_vector_type(2)));
typedef float v8f __attribute__((ext_vector_type(8)));

// =====================================================================
// Kernel 1: U = X(T x 8) @ W_in^T(8 x R) via V_WMMA_F32_16X16X4_F32.
// One wave -> one 16x16 tile of U, two K=4 WMMAs (K total = 8), exact f32.
// Lane layouts per CDNA5 ISA 7.12.2 (A: lanes=M, half-wave splits K;
// B: lanes=N likewise; D: VGPR i -> M=i / i+8, N=lane&15).
// =====================================================================
__global__ __launch_bounds__(256) void esn_uproj_wmma(
    const float* __restrict__ X,      // (T, 8)
    const float* __restrict__ Win,    // (R, 8)
    float* __restrict__ U)            // (T, R)
{
  const int wave   = (int)((blockIdx.x * blockDim.x + threadIdx.x) >> 5);
  const int lane   = (int)(threadIdx.x & 31);
  const int tilesN = RDIM / 16;                // 64
  const int m0 = (wave / tilesN) * 16;
  const int n0 = (wave % tilesN) * 16;
  const int half = lane >> 4;
  const int mr   = lane & 15;

  v8f c = {0.f,0.f,0.f,0.f,0.f,0.f,0.f,0.f};
#pragma unroll
  for (int kb = 0; kb < IDIM; kb += 4) {
    const int k = kb + 2 * half;
    v2f a, b;
    a.x = X  [(size_t)(m0 + mr) * IDIM + k];
    a.y = X  [(size_t)(m0 + mr) * IDIM + k + 1];
    b.x = Win[(size_t)(n0 + mr) * IDIM + k];
    b.y = Win[(size_t)(n0 + mr) * IDIM + k + 1];
    c = __builtin_amdgcn_wmma_f32_16x16x4_f32(false, a, false, b, (short)0, c,
                                              false, false);
  }
#pragma unroll
  for (int i = 0; i < 8; ++i)
    U[(size_t)(m0 + i + 8 * half) * RDIM + (n0 + mr)] = c[i];
}

// =====================================================================
// Kernel 2: bottom I rows of output: out[R+i][t] = x[t][i]
// =====================================================================
__global__ __launch_bounds__(256) void esn_xrows(
    const float* __restrict__ X, float* __restrict__ out)
{
  const unsigned gid = blockIdx.x * blockDim.x + threadIdx.x;   // 0 .. 8*T-1
  const unsigned ii = gid >> 16;
  const unsigned t  = gid & (TSTEPS - 1);
  out[(size_t)(RDIM + ii) * TSTEPS + t] = X[(size_t)t * IDIM + ii];
}

// =====================================================================
// Kernel 3: persistent serial scan. 32 WGs x 1024 threads.
// Each WG keeps a 32x1024 W_res tile in LDS (128 KB), swizzled as
// float4 Wt[(k>>2)*32 + row] so ds_load_b128 is conflict-free across
// lanes (rows) and the state chunk is a wave-uniform LDS broadcast.
// Cross-WG sync: per-WG padded flag array (release store + parallel
// relaxed DEV-scope polling); state double-buffered through L2.
// =====================================================================
__global__ __launch_bounds__(SCAN_THREADS, 1) void esn_scan_kernel(
    const float* __restrict__ inputs,   // (T, 8)
    const float* __restrict__ W_in,     // (R, 8)
    const float* __restrict__ W_res,    // (R, R)
    const float* __restrict__ state0,   // (R)
    const float* __restrict__ U,        // (T, R) or unused
    int useU,
    float* __restrict__ out,            // (R+I, T)
    unsigned int* __restrict__ flags,   // ws, NWG padded flags (zeroed/launch)
    float* __restrict__ s_buf)          // ws, 2*R floats (double buffer)
{
  extern __shared__ float smem[];
  float* Wt     = smem;                                  // 32*1024
  float* s_sh   = Wt + ROWS_PER_WG * RDIM;               // 1024
  float* red    = s_sh + RDIM;                           // 32*32
  float* Win_sh = red + KSEG * ROWS_PER_WG;              // 32*8

  const int tid   = (int)threadIdx.x;
  const int wg    = (int)blockIdx.x;                     // 0..31
  const int row   = tid & (ROWS_PER_WG - 1);
  const int seg   = tid >> 5;                            // 0..31 (wave-uniform)
  const int grow0 = wg * ROWS_PER_WG;

  // ---- one-time: stage W_res tile into LDS (coalesced global reads) ----
#pragma unroll 4
  for (int it = 0; it < (ROWS_PER_WG * RDIM) / SCAN_THREADS; ++it) {
    const int idx = tid + it * SCAN_THREADS;
    const int r = idx >> 10;
    const int k = idx & (RDIM - 1);
    Wt[((k >> 2) * ROWS_PER_WG + r) * 4 + (k & 3)] =
        W_res[(size_t)(grow0 + r) * RDIM + k];
  }
  if (tid < ROWS_PER_WG * IDIM)
    Win_sh[tid] = W_in[(size_t)grow0 * IDIM + tid];
  s_sh[tid] = state0[tid];
  __syncthreads();

  float* sb0 = s_buf;
  float* sb1 = s_buf + RDIM;

  for (int t = 0; t < TSTEPS; ++t) {
    float* sb_cur = (t & 1) ? sb1 : sb0;

    // ---- input projection first: independent of state, hides latency ----
    float proj = 0.f;
    if (tid < ROWS_PER_WG) {
      if (useU) {
        proj = U[(size_t)t * RDIM + grow0 + tid];
        if (t + PF_DIST < TSTEPS)   // keep our 128B slice streaming into L2
          __builtin_prefetch(&U[(size_t)(t + PF_DIST) * RDIM + grow0 + tid], 0, 3);
      } else {
        const float* x = inputs + (size_t)t * IDIM;
#pragma unroll
        for (int i = 0; i < IDIM; ++i)
          proj = fmaf(Win_sh[tid * IDIM + i], x[i], proj);
      }
    }

    // ---- partial dot over k in [seg*32, seg*32+32): 8 b128 LDS loads ----
    const float4* Wt4 = (const float4*)Wt;
    const float4* s4p = (const float4*)s_sh;
    const int blk0 = seg * (KPER / 4);
    float acc = 0.f;
#pragma unroll
    for (int b = 0; b < KPER / 4; ++b) {
      const float4 w = Wt4[(blk0 + b) * ROWS_PER_WG + row];
      const float4 s = s4p[blk0 + b];                    // broadcast
      acc = fmaf(w.x, s.x, acc);
      acc = fmaf(w.y, s.y, acc);
      acc = fmaf(w.z, s.z, acc);
      acc = fmaf(w.w, s.w, acc);
    }
    red[seg * ROWS_PER_WG + row] = acc;
    __syncthreads();

    // ---- reduce KSEG partials per row, tanh, publish slice ----
    if (tid < ROWS_PER_WG) {
      float tot = proj;
#pragma unroll
      for (int i = 0; i < KSEG; ++i) tot += red[i * ROWS_PER_WG + tid];
      const float sv = tanhf(tot);
      out[(size_t)(grow0 + tid) * TSTEPS + t] = sv;
      __hip_atomic_store(&sb_cur[grow0 + tid], sv, __ATOMIC_RELAXED,
                         __HIP_MEMORY_SCOPE_AGENT);
      __threadfence();   // slice stores reach the L2 coherence point
    }
    __syncthreads();

    // ---- device-wide barrier: 1 release store, NWG parallel spinners ----
    if (tid == 0)
      __hip_atomic_store(&flags[wg * FLAG_STRIDE], (unsigned)(t + 1),
                         __ATOMIC_RELEASE, __HIP_MEMORY_SCOPE_AGENT);
    if (tid < NWG) {
      while (__hip_atomic_load(&flags[tid * FLAG_STRIDE], __ATOMIC_RELAXED,
                               __HIP_MEMORY_SCOPE_AGENT) < (unsigned)(t + 1)) {
      }
    }
    __syncthreads();

    // ---- pull full new state (DEV-scope loads bypass stale WGP$) ----
    s_sh[tid] = __hip_atomic_load(&sb_cur[tid], __ATOMIC_RELAXED,
                                  __HIP_MEMORY_SCOPE_AGENT);
    __syncthreads();
  }
}

// =====================================================================
extern "C" void kernel_launch(void* const* d_in, const int* in_sizes, int n_in,
                              void* d_out, int out_size, void* d_ws,
                              size_t ws_size, hipStream_t stream) {
  const float* inputs = (const float*)d_in[0];   // (65536, 8)
  const float* W_in   = (const float*)d_in[1];   // (1024, 8)
  const float* W_res  = (const float*)d_in[2];   // (1024, 1024)
  const float* state0 = (const float*)d_in[3];   // (1024,)
  float* out = (float*)d_out;                    // (1032, 65536)

  unsigned char* ws = (unsigned char*)d_ws;
  unsigned int* flags = (unsigned int*)ws;                   // @0, padded
  float* s_buf = (float*)(ws + WS_SBUF_OFF);                 // 2*1024 f32
  float* U     = (float*)(ws + WS_U_OFF);                    // (T, R) f32
  const size_t u_bytes = (size_t)TSTEPS * RDIM * sizeof(float);
  const int useU = (ws_size >= (size_t)WS_U_OFF + u_bytes) ? 1 : 0;

  // reset barrier flags every call (graph-capture-legal memset node)
  hipMemsetAsync(ws, 0, WS_SBUF_OFF, stream);

  if (useU) {
    const int tiles = (TSTEPS / 16) * (RDIM / 16);           // 262144
    esn_uproj_wmma<<<tiles / 8, 256, 0, stream>>>(inputs, W_in, U);
  }

  esn_xrows<<<(IDIM * TSTEPS) / 256, 256, 0, stream>>>(inputs, out);

  esn_scan_kernel<<<NWG, SCAN_THREADS, SMEM_BYTES, stream>>>(
      inputs, W_in, W_res, state0, U, useU, out, flags, s_buf);
}